// PopularitySampler_19576460935745
// MI455X (gfx1250) — compile-verified
//
#include <hip/hip_runtime.h>
#include <stdint.h>

#define TOPN 8192   // 32KB LDS top-level CDF table

// ---------- Pass 1: per-chunk (1024 elems) partial sums ----------
__global__ void __launch_bounds__(256)
k_partial(const float* __restrict__ f, float* __restrict__ partials, int n) {
  __shared__ float red[256];
  int b = blockIdx.x, t = threadIdx.x;
  int base = (b << 10) + (t << 2);
  float s = 0.f;
  if (base + 3 < n) {                       // fast path: unguarded b128 load
    float4 v = *(const float4*)(f + base);
    s = (v.x + v.y) + (v.z + v.w);
  } else {                                  // tail block only
    for (int k = 0; k < 4; ++k) { int g = base + k; if (g < n) s += f[g]; }
  }
  red[t] = s;
  __syncthreads();
  for (int off = 128; off > 0; off >>= 1) {
    if (t < off) red[t] += red[t + off];
    __syncthreads();
  }
  if (t == 0) partials[b] = red[0];
}

// ---------- Pass 2: single-block inclusive scan of partials (<=2048) ----------
__global__ void __launch_bounds__(1024)
k_scan(float* __restrict__ partials, float* __restrict__ total, int nb) {
  __shared__ float buf[2][2048];
  int t = threadIdx.x;
  for (int i = t; i < 2048; i += 1024) buf[0][i] = (i < nb) ? partials[i] : 0.f;
  __syncthreads();
  int src = 0;
  for (int off = 1; off < 2048; off <<= 1) {
    int dst = src ^ 1;
    for (int i = t; i < 2048; i += 1024)
      buf[dst][i] = (i >= off) ? (buf[src][i] + buf[src][i - off]) : buf[src][i];
    __syncthreads();
    src = dst;
  }
  for (int i = t; i < 2048; i += 1024)
    if (i < nb) partials[i] = buf[src][i];
  if (t == 0) *total = buf[src][nb - 1];
}

// ---------- Pass 3: per-chunk inclusive cumsum + normalize ----------
__global__ void __launch_bounds__(256)
k_cumsum(const float* __restrict__ f, const float* __restrict__ scanned,
         const float* __restrict__ total, float* __restrict__ cum, int n) {
  __shared__ float sc[256];
  int b = blockIdx.x, t = threadIdx.x;
  int base = (b << 10) + (t << 2);
  float v0 = 0.f, v1 = 0.f, v2 = 0.f, v3 = 0.f;
  bool full = (base + 3 < n);
  if (full) {
    float4 v = *(const float4*)(f + base);
    v0 = v.x; v1 = v.y; v2 = v.z; v3 = v.w;
  } else {
    if (base + 0 < n) v0 = f[base + 0];
    if (base + 1 < n) v1 = f[base + 1];
    if (base + 2 < n) v2 = f[base + 2];
    if (base + 3 < n) v3 = f[base + 3];
  }
  float p0 = v0, p1 = p0 + v1, p2 = p1 + v2, p3 = p2 + v3;
  sc[t] = p3;
  __syncthreads();
#pragma unroll
  for (int off = 1; off < 256; off <<= 1) {
    float x = (t >= off) ? sc[t - off] : 0.f;
    __syncthreads();
    sc[t] += x;
    __syncthreads();
  }
  float ex = sc[t] - p3 + ((b > 0) ? scanned[b - 1] : 0.f);
  float inv = 1.f / (*total);
  if (full) {
    float4 o = make_float4((ex + p0) * inv, (ex + p1) * inv,
                           (ex + p2) * inv, (ex + p3) * inv);
    *(float4*)(cum + base) = o;
  } else {
    float r[4] = {p0, p1, p2, p3};
    for (int k = 0; k < 4; ++k) { int g = base + k; if (g < n) cum[g] = (ex + r[k]) * inv; }
  }
}

// ---------- Pass 4: build the TOPN-entry top-level CDF table ----------
__global__ void __launch_bounds__(256)
k_top(const float* __restrict__ cum, float* __restrict__ top, int n, int stride) {
  int j = blockIdx.x * 256 + threadIdx.x;
  if (j < TOPN) {
    long long e = (long long)(j + 1) * (long long)stride;
    if (e > n) e = n;
    top[j] = cum[e - 1];
  }
}

// ---------- two-level upper_bound ----------
__device__ __forceinline__ int search2(float uv, const float* s_top,
                                       const float* __restrict__ cum,
                                       int n, int stride) {
  int lo = 0, hi = TOPN;                     // 13 LDS steps
  while (lo < hi) {
    int mid = (lo + hi) >> 1;
    if (s_top[mid] > uv) hi = mid; else lo = mid + 1;
  }
  int idx;
  if (lo >= TOPN) {
    idx = n;
  } else {
    int s = lo * stride;                     // <=8 L2 steps (segment <= 245)
    int e = s + stride; if (e > n) e = n;
    while (s < e) {
      int mid = (s + e) >> 1;
      if (cum[mid] > uv) e = mid; else s = mid + 1;
    }
    idx = s;
  }
  if (idx > n - 1) idx = n - 1;
  return idx;
}

// ---------- Pass 5: two-level searchsorted + gather ----------
__global__ void __launch_bounds__(256)
k_lookup(const float* __restrict__ u, const float* __restrict__ cum,
         const float* __restrict__ top, const float* __restrict__ freq,
         float* __restrict__ out_idx, float* __restrict__ out_prob,
         int m, int n, int stride) {
  __shared__ float s_top[TOPN];
  int t = threadIdx.x;

  // CDNA5 async global->LDS copy of the 32KB top table (ASYNCcnt-tracked,
  // bypasses VGPRs). 256 lanes x 16B x 8 rounds = 32KB.
#pragma unroll
  for (int r = 0; r < TOPN / (256 * 4); ++r) {
    int c = (r << 8) + t;                                   // 16-byte chunk id
    unsigned lds = (unsigned)(uintptr_t)(&s_top[c << 2]);   // low 32b = LDS offset
    const float* g = top + (c << 2);
    asm volatile("global_load_async_to_lds_b128 %0, %1, off"
                 :: "v"(lds), "v"(g) : "memory");
  }
  asm volatile("s_wait_asynccnt 0x0" ::: "memory");
  __syncthreads();

  // 2 tiles x 4 contiguous items per thread: b128 u-load, two b128 stores
#pragma unroll
  for (int half = 0; half < 2; ++half) {
    int i0 = blockIdx.x * 2048 + (half << 10) + (t << 2);
    if (i0 + 3 < m) {
      float4 uv = *(const float4*)(u + i0);
      int i_x = search2(uv.x, s_top, cum, n, stride);   // 4 independent chains:
      int i_y = search2(uv.y, s_top, cum, n, stride);   // MLP inside the wave
      int i_z = search2(uv.z, s_top, cum, n, stride);
      int i_w = search2(uv.w, s_top, cum, n, stride);
      float4 oi = make_float4((float)(i_x + 1), (float)(i_y + 1),
                              (float)(i_z + 1), (float)(i_w + 1));
      float4 op = make_float4(__logf(freq[i_x] + 1.0f), __logf(freq[i_y] + 1.0f),
                              __logf(freq[i_z] + 1.0f), __logf(freq[i_w] + 1.0f));
      *(float4*)(out_idx  + i0) = oi;
      *(float4*)(out_prob + i0) = op;
    } else {
      for (int k = 0; k < 4; ++k) {
        int i = i0 + k;
        if (i < m) {
          int idx = search2(u[i], s_top, cum, n, stride);
          out_idx[i]  = (float)(idx + 1);
          out_prob[i] = __logf(freq[idx] + 1.0f);
        }
      }
    }
  }
}

extern "C" void kernel_launch(void* const* d_in, const int* in_sizes, int n_in,
                              void* d_out, int out_size, void* d_ws, size_t ws_size,
                              hipStream_t stream) {
  // inputs: [0]=trg_seq(int,L) [1]=k [2]=user [3]=loc2freq(float,N) [4]=u(float,L*K)
  const float* loc2freq = (const float*)d_in[3];
  const float* u        = (const float*)d_in[4];
  int n = in_sizes[3];
  int m = in_sizes[4];

  float* ws       = (float*)d_ws;
  float* cum      = ws;                                   // n floats
  float* partials = ws + n;                               // 2048 floats
  float* total    = partials + 2048;                      // 1 float (+pad)
  float* top      = (float*)(((uintptr_t)(total + 1) + 15) & ~(uintptr_t)15); // 16B aligned

  int nb     = (n + 1023) >> 10;
  int stride = (n + TOPN - 1) / TOPN;

  k_partial<<<nb, 256, 0, stream>>>(loc2freq, partials, n);
  k_scan<<<1, 1024, 0, stream>>>(partials, total, nb);
  k_cumsum<<<nb, 256, 0, stream>>>(loc2freq, partials, total, cum, n);
  k_top<<<(TOPN + 255) / 256, 256, 0, stream>>>(cum, top, n, stride);

  float* out_idx  = (float*)d_out;
  float* out_prob = out_idx + m;
  int blocks = (m + 2047) / 2048;
  k_lookup<<<blocks, 256, 0, stream>>>(u, cum, top, loc2freq,
                                       out_idx, out_prob, m, n, stride);
}